// Net_46755013984382
// MI455X (gfx1250) — compile-verified
//
#include <hip/hip_runtime.h>
#include <math.h>

typedef __attribute__((ext_vector_type(2))) float v2f;
typedef __attribute__((ext_vector_type(8))) float v8f;

#define KSPLINE 5
#define KKSLOTS 25
#define NUM_GRAPHS 64
#define NNODES 30000
#define NEDGES 600000

// ---------------------------------------------------------------------------
// Pack Wt column-major: Wt[(k*O + o) * I + i] = W[k, i, o]
// GEMM B operand becomes Bt[col * KDIM + k]; each lane's 2-element K fragment
// is a contiguous, 8B-aligned float2 -> single global_load_b64.
// ---------------------------------------------------------------------------
__global__ void pack_w_kernel(const float* __restrict__ W, float* __restrict__ Wt,
                              int I, int O) {
  int idx = blockIdx.x * blockDim.x + threadIdx.x;
  int total = KKSLOTS * I * O;
  if (idx >= total) return;
  int o = idx % O;
  int t = idx / O;
  int i = t % I;
  int k = t / I;
  Wt[(k * O + o) * I + i] = W[idx];  // idx == k*I*O + i*O + o
}

// Bt[o*I + i] = B[i*O + o]  (root matrix -> column-major)
__global__ void transpose_b_kernel(const float* __restrict__ B, float* __restrict__ Bt,
                                   int I, int O) {
  int idx = blockIdx.x * blockDim.x + threadIdx.x;
  if (idx >= I * O) return;
  int o = idx % O;
  int i = idx / O;
  Bt[o * I + i] = B[idx];
}

// ---------------------------------------------------------------------------
// C[M x Ccols] = A[M x KDIM] * B[KDIM x Ccols], fp32, exact accumulate via
// V_WMMA_F32_16X16X4_F32.  B supplied column-major as Bt[Ccols x KDIM].
// Block = 128 threads = 4 waves; one 16-row A stripe in LDS shared by all
// waves; each wave owns two 16-col tiles (A-fragment reuse, 2 WMMA per step).
// M multiple of 16 (30000 is). Out-of-range columns are clamped, not
// predicated: their lanes compute garbage that is simply never stored, so
// EXEC stays all-1s through every WMMA (ISA requirement) and no exec-mask
// branches appear in the inner loop.
// ---------------------------------------------------------------------------
template <int KDIM>
__global__ void wmma_gemm_f32(const float* __restrict__ A, const float* __restrict__ Bt,
                              float* __restrict__ C, int Ccols) {
  __shared__ float lds_a[16 * KDIM];

  const int tid  = threadIdx.x;
  const int wave = tid >> 5;
  const int lane = tid & 31;
  const int hh   = lane >> 4;   // wave half: selects K pair (A/B) and M half (D)
  const int l16  = lane & 15;
  const int row0 = blockIdx.y * 16;
  const int col0 = blockIdx.x * 128 + wave * 32;

  // A stripe (16 rows x KDIM) is contiguous in memory; copy with b128s.
  constexpr int TOT = 16 * KDIM;
  const float* Astripe = A + (size_t)row0 * KDIM;  // 256B aligned
#pragma unroll
  for (int idx = tid * 4; idx < TOT; idx += 128 * 4)
    *(float4*)(lds_a + idx) = *(const float4*)(Astripe + idx);
  __syncthreads();

  const int n0  = col0 + l16;
  const int n1  = n0 + 16;
  const int nc0 = min(n0, Ccols - 1);
  const int nc1 = min(n1, Ccols - 1);

  v8f acc0 = {}, acc1 = {};
#pragma unroll
  for (int k0 = 0; k0 < KDIM; k0 += 4) {
    const int ka = k0 + 2 * hh;
    v2f a, b0, b1;
    a.x = lds_a[l16 * KDIM + ka];
    a.y = lds_a[l16 * KDIM + ka + 1];
    b0 = *(const v2f*)(Bt + nc0 * KDIM + ka);   // contiguous 8B, aligned
    b1 = *(const v2f*)(Bt + nc1 * KDIM + ka);
    acc0 = __builtin_amdgcn_wmma_f32_16x16x4_f32(false, a, false, b0,
                                                 (short)0, acc0, false, false);
    acc1 = __builtin_amdgcn_wmma_f32_16x16x4_f32(false, a, false, b1,
                                                 (short)0, acc1, false, false);
  }

  const int mbase = row0 + 8 * hh;  // D layout: VGPR r -> row 8*hh + r
  if (n0 < Ccols) {
#pragma unroll
    for (int r = 0; r < 8; ++r)
      C[(size_t)(mbase + r) * Ccols + n0] = acc0[r];
  }
  if (n1 < Ccols) {
#pragma unroll
    for (int r = 0; r < 8; ++r)
      C[(size_t)(mbase + r) * Ccols + n1] = acc1[r];
  }
}

// ---------------------------------------------------------------------------
// In-degree (same for all three layers)
// ---------------------------------------------------------------------------
__global__ void deg_kernel(const int* __restrict__ dst, float* __restrict__ deg) {
  int e = blockIdx.x * blockDim.x + threadIdx.x;
  if (e < NEDGES) atomicAdd(&deg[dst[e]], 1.0f);
}

// ---------------------------------------------------------------------------
// Edge message: one 32-lane wave per edge (blockDim = 32 x 8). Lanes stride
// the feature dim so the 4 gathered Xk rows (contiguous O-float rows) and the
// atomic scatter into agg[dst] are coalesced across the wave.
// ---------------------------------------------------------------------------
__global__ void edge_message_kernel(const int* __restrict__ src, const int* __restrict__ dst,
                                    const float* __restrict__ pseudo,
                                    const float* __restrict__ Xk,
                                    float* __restrict__ agg, int O) {
  int e = blockIdx.x * blockDim.y + threadIdx.y;
  if (e >= NEDGES) return;
  const int s = src[e];
  const int d = dst[e];

  const float2 ps = ((const float2*)pseudo)[e];
  const float u0 = ps.x * (float)(KSPLINE - 1);
  const float u1 = ps.y * (float)(KSPLINE - 1);
  const float fl0 = floorf(u0), fl1 = floorf(u1);
  const float fr0 = u0 - fl0, fr1 = u1 - fl1;
  const int b0 = (int)fl0, b1 = (int)fl1;

  float w[4];
  int   base[4];
  int c = 0;
#pragma unroll
  for (int s0 = 0; s0 < 2; ++s0) {
    const float w0 = s0 ? fr0 : 1.0f - fr0;
    const int   i0 = min(max(b0 + s0, 0), KSPLINE - 1);
#pragma unroll
    for (int s1 = 0; s1 < 2; ++s1) {
      const float w1 = s1 ? fr1 : 1.0f - fr1;
      const int   i1 = min(max(b1 + s1, 0), KSPLINE - 1);
      w[c]    = w0 * w1;
      base[c] = (s * KKSLOTS + i0 * KSPLINE + i1) * O;
      ++c;
    }
  }

  for (int f = threadIdx.x; f < O; f += 32) {
    float m = w[0] * Xk[base[0] + f] + w[1] * Xk[base[1] + f] +
              w[2] * Xk[base[2] + f] + w[3] * Xk[base[3] + f];
    atomicAdd(&agg[d * O + f], m);
  }
}

// ---------------------------------------------------------------------------
// h = elu(agg/deg + x@root + bias)
// ---------------------------------------------------------------------------
__global__ void finalize_kernel(const float* __restrict__ agg, const float* __restrict__ deg,
                                const float* __restrict__ xroot, const float* __restrict__ bias,
                                float* __restrict__ h, int O) {
  int idx = blockIdx.x * blockDim.x + threadIdx.x;
  int total = NNODES * O;
  if (idx >= total) return;
  int n = idx / O;
  int f = idx - n * O;
  float v = agg[idx] / fmaxf(deg[n], 1.0f) + xroot[idx] + bias[f];
  h[idx] = (v > 0.0f) ? v : expm1f(v);
}

// ---------------------------------------------------------------------------
// Batch mean accumulation (final layer, O = 124)
// ---------------------------------------------------------------------------
__global__ void batch_sum_kernel(const float* __restrict__ h, const int* __restrict__ batch,
                                 float* __restrict__ gsum, float* __restrict__ gcnt) {
  int idx = blockIdx.x * blockDim.x + threadIdx.x;
  int total = NNODES * 124;
  if (idx >= total) return;
  int n = idx / 124;
  int f = idx - n * 124;
  int g = batch[n];
  atomicAdd(&gsum[g * 124 + f], h[idx]);
  if (f == 0) atomicAdd(&gcnt[g], 1.0f);
}

// ---------------------------------------------------------------------------
// Head: mean -> FC(124->30) -> log_softmax.  64 graphs, one thread each.
// ---------------------------------------------------------------------------
__global__ void head_kernel(const float* __restrict__ gsum, const float* __restrict__ gcnt,
                            const float* __restrict__ fcw, const float* __restrict__ fcb,
                            float* __restrict__ out) {
  int g = threadIdx.x;
  if (g >= NUM_GRAPHS) return;
  float inv = 1.0f / fmaxf(gcnt[g], 1.0f);
  float logits[30];
#pragma unroll 1
  for (int c = 0; c < 30; ++c) {
    float sacc = fcb[c];
    for (int i = 0; i < 124; ++i)
      sacc += (gsum[g * 124 + i] * inv) * fcw[i * 30 + c];
    logits[c] = sacc;
  }
  float mx = logits[0];
  for (int c = 1; c < 30; ++c) mx = fmaxf(mx, logits[c]);
  float se = 0.0f;
  for (int c = 0; c < 30; ++c) se += expf(logits[c] - mx);
  float lse = logf(se);
  for (int c = 0; c < 30; ++c) out[g * 30 + c] = logits[c] - mx - lse;
}

// ---------------------------------------------------------------------------
extern "C" void kernel_launch(void* const* d_in, const int* in_sizes, int n_in,
                              void* d_out, int out_size, void* d_ws, size_t ws_size,
                              hipStream_t stream) {
  const float* x      = (const float*)d_in[0];
  const int*   ei     = (const int*)d_in[1];
  const float* pseudo = (const float*)d_in[2];
  const int*   batch  = (const int*)d_in[3];
  const float* W1 = (const float*)d_in[4];
  const float* r1 = (const float*)d_in[5];
  const float* b1 = (const float*)d_in[6];
  const float* W2 = (const float*)d_in[7];
  const float* r2 = (const float*)d_in[8];
  const float* b2 = (const float*)d_in[9];
  const float* W3 = (const float*)d_in[10];
  const float* r3 = (const float*)d_in[11];
  const float* b3 = (const float*)d_in[12];
  const float* fcw = (const float*)d_in[13];
  const float* fcb = (const float*)d_in[14];
  float* out = (float*)d_out;

  const int* src = ei;
  const int* dst = ei + NEDGES;

  // Workspace carve (bump allocator, 256B aligned)
  char* wp = (char*)d_ws;
  auto carve = [&](size_t bytes) -> float* {
    float* p = (float*)wp;
    wp += (bytes + 255) & ~(size_t)255;
    return p;
  };
  float* Xk   = carve((size_t)NNODES * KKSLOTS * 124 * 4);  // 372 MB (layer-3 max)
  float* Wt   = carve((size_t)64 * KKSLOTS * 124 * 4);      // packed spline weights
  float* Rt   = carve((size_t)64 * 124 * 4);                // packed root weights
  float* Xr   = carve((size_t)NNODES * 124 * 4);
  float* agg  = carve((size_t)NNODES * 124 * 4);
  float* deg  = carve((size_t)NNODES * 4);
  float* h1   = carve((size_t)NNODES * 32 * 4);
  float* h2   = carve((size_t)NNODES * 64 * 4);
  float* h3   = carve((size_t)NNODES * 124 * 4);
  float* gsum = carve((size_t)NUM_GRAPHS * 124 * 4);
  float* gcnt = carve((size_t)NUM_GRAPHS * 4);

  // In-degree (shared by all layers)
  hipMemsetAsync(deg, 0, (size_t)NNODES * 4, stream);
  deg_kernel<<<(NEDGES + 255) / 256, 256, 0, stream>>>(dst, deg);

  auto launch_gemm = [&](const float* A, const float* Bt, float* C, int I, int Ccols) {
    dim3 g((Ccols + 127) / 128, NNODES / 16);
    if (I == 8)
      wmma_gemm_f32<8><<<g, 128, 0, stream>>>(A, Bt, C, Ccols);
    else if (I == 32)
      wmma_gemm_f32<32><<<g, 128, 0, stream>>>(A, Bt, C, Ccols);
    else
      wmma_gemm_f32<64><<<g, 128, 0, stream>>>(A, Bt, C, Ccols);
  };

  auto layer = [&](const float* hin, int I, int O, const float* W,
                   const float* root, const float* bias, float* hout) {
    const int Ck = KKSLOTS * O;
    pack_w_kernel<<<(KKSLOTS * I * O + 255) / 256, 256, 0, stream>>>(W, Wt, I, O);
    transpose_b_kernel<<<(I * O + 255) / 256, 256, 0, stream>>>(root, Rt, I, O);
    launch_gemm(hin, Wt, Xk, I, Ck);
    launch_gemm(hin, Rt, Xr, I, O);
    hipMemsetAsync(agg, 0, (size_t)NNODES * O * 4, stream);
    dim3 eb(32, 8);
    edge_message_kernel<<<(NEDGES + 7) / 8, eb, 0, stream>>>(src, dst, pseudo, Xk, agg, O);
    int tot = NNODES * O;
    finalize_kernel<<<(tot + 255) / 256, 256, 0, stream>>>(agg, deg, Xr, bias, hout, O);
  };

  layer(x,  8,   32, W1, r1, b1, h1);
  layer(h1, 32,  64, W2, r2, b2, h2);
  layer(h2, 64, 124, W3, r3, b3, h3);

  hipMemsetAsync(gsum, 0, (size_t)NUM_GRAPHS * 124 * 4, stream);
  hipMemsetAsync(gcnt, 0, (size_t)NUM_GRAPHS * 4, stream);
  int tot = NNODES * 124;
  batch_sum_kernel<<<(tot + 255) / 256, 256, 0, stream>>>(h3, batch, gsum, gcnt);
  head_kernel<<<1, 64, 0, stream>>>(gsum, gcnt, fcw, fcb, out);
}